// EmbedDNF_7456063226495
// MI455X (gfx1250) — compile-verified
//
#include <hip/hip_runtime.h>
#include <hip/hip_bf16.h>
#include <stdint.h>

// Problem constants (from reference): B, IN_F, HID_F, OUT_F, E
#define BB   128
#define INF  256
#define HID  512
#define OUTF 128
#define EE   8

typedef float v4f __attribute__((ext_vector_type(4)));

// ---------------- CDNA5 async global->LDS staging ----------------
// Stages 64KB (contiguous) from global to LDS using GVS-mode async loads:
//   each lane copies 16 B from global saddr+vaddr+offset to LDS at vdst+offset
// per-lane: vdst = lds_base+tid*16, vaddr = tid*16; 32 chunks via literal offset:.
// Requires nthreads == 128 (chunk stride = 128*16 = 2048 B).
__device__ __forceinline__ void stage_64kb_async(void* lds_base, const void* gsrc, int tid) {
#if defined(__gfx1250__)
    uint32_t lds_off = (uint32_t)(uintptr_t)lds_base + (uint32_t)tid * 16u;
    uint32_t g_off   = (uint32_t)tid * 16u;
    asm volatile(
        "global_load_async_to_lds_b128 %0, %1, %2 offset:0\n\t"
        "global_load_async_to_lds_b128 %0, %1, %2 offset:2048\n\t"
        "global_load_async_to_lds_b128 %0, %1, %2 offset:4096\n\t"
        "global_load_async_to_lds_b128 %0, %1, %2 offset:6144\n\t"
        "global_load_async_to_lds_b128 %0, %1, %2 offset:8192\n\t"
        "global_load_async_to_lds_b128 %0, %1, %2 offset:10240\n\t"
        "global_load_async_to_lds_b128 %0, %1, %2 offset:12288\n\t"
        "global_load_async_to_lds_b128 %0, %1, %2 offset:14336\n\t"
        "global_load_async_to_lds_b128 %0, %1, %2 offset:16384\n\t"
        "global_load_async_to_lds_b128 %0, %1, %2 offset:18432\n\t"
        "global_load_async_to_lds_b128 %0, %1, %2 offset:20480\n\t"
        "global_load_async_to_lds_b128 %0, %1, %2 offset:22528\n\t"
        "global_load_async_to_lds_b128 %0, %1, %2 offset:24576\n\t"
        "global_load_async_to_lds_b128 %0, %1, %2 offset:26624\n\t"
        "global_load_async_to_lds_b128 %0, %1, %2 offset:28672\n\t"
        "global_load_async_to_lds_b128 %0, %1, %2 offset:30720\n\t"
        "global_load_async_to_lds_b128 %0, %1, %2 offset:32768\n\t"
        "global_load_async_to_lds_b128 %0, %1, %2 offset:34816\n\t"
        "global_load_async_to_lds_b128 %0, %1, %2 offset:36864\n\t"
        "global_load_async_to_lds_b128 %0, %1, %2 offset:38912\n\t"
        "global_load_async_to_lds_b128 %0, %1, %2 offset:40960\n\t"
        "global_load_async_to_lds_b128 %0, %1, %2 offset:43008\n\t"
        "global_load_async_to_lds_b128 %0, %1, %2 offset:45056\n\t"
        "global_load_async_to_lds_b128 %0, %1, %2 offset:47104\n\t"
        "global_load_async_to_lds_b128 %0, %1, %2 offset:49152\n\t"
        "global_load_async_to_lds_b128 %0, %1, %2 offset:51200\n\t"
        "global_load_async_to_lds_b128 %0, %1, %2 offset:53248\n\t"
        "global_load_async_to_lds_b128 %0, %1, %2 offset:55296\n\t"
        "global_load_async_to_lds_b128 %0, %1, %2 offset:57344\n\t"
        "global_load_async_to_lds_b128 %0, %1, %2 offset:59392\n\t"
        "global_load_async_to_lds_b128 %0, %1, %2 offset:61440\n\t"
        "global_load_async_to_lds_b128 %0, %1, %2 offset:63488"
        :: "v"(lds_off), "v"(g_off), "s"(gsrc)
        : "memory");
    asm volatile("s_wait_asynccnt 0x0" ::: "memory");
#else
    // generic fallback (never used on gfx1250)
    for (int k = 0; k < 32; ++k) {
        int byte_off = (tid + k * 128) * 16;
        *(float4*)((char*)lds_base + byte_off) = *(const float4*)((const char*)gsrc + byte_off);
    }
#endif
}

// ---------------- Stage 1 ----------------
// h[b,hh,e] = prod_i ( C[i,hh,e] + A[i,hh,e] * x[b,i,e] )
//   A = w*(2s-1), C = 1 - w*s   (on the fly, amortized over BT1 b's)
// Block: 128 threads (4 waves). lane -> (h_local in [0,64), e-quad in [0,2)).
// All global/LDS data accesses are 16B (b128); vector math contracts to v_pk_fma_f32.
#define HT1 64
#define BT1 8

__global__ __launch_bounds__(128) void dnf_stage1(const float* __restrict__ x,
                                                  const float* __restrict__ cw,
                                                  const float* __restrict__ cs,
                                                  float* __restrict__ hbuf) {
    __shared__ float xs[BT1 * INF * EE];   // 16384 floats = 64 KB

    const int tid = threadIdx.x;
    const int eq  = tid & 1;          // e-quad index, e = 4*eq
    const int hl  = tid >> 1;         // 0..63
    const int h   = blockIdx.x * HT1 + hl;
    const int b0  = blockIdx.y * BT1;

    // async-stage contiguous 64KB x-tile: x[b0 .. b0+7, :, :]
    stage_64kb_async(xs, x + (size_t)b0 * INF * EE, tid);
    __syncthreads();

    const size_t colOff = (size_t)h * EE + 4 * eq;   // (h, e-quad) column offset
    const float* wp = cw + colOff;
    const float* sp = cs + colOff;

    v4f acc[BT1];
    #pragma unroll
    for (int b = 0; b < BT1; ++b) acc[b] = (v4f){1.0f, 1.0f, 1.0f, 1.0f};

    #pragma unroll 2
    for (int i = 0; i < INF; ++i) {
        const size_t row = (size_t)i * (HID * EE);
        v4f w4 = *(const v4f*)(wp + row);            // global_load_b128
        v4f s4 = *(const v4f*)(sp + row);            // global_load_b128

        v4f A = w4 * (2.0f * s4 - 1.0f);             // A = w*(2s-1)
        v4f C = 1.0f - w4 * s4;                      // C = 1 - w*s  (contracts to fma)

        #pragma unroll
        for (int b = 0; b < BT1; ++b) {
            v4f xv = *(const v4f*)&xs[(b * INF + i) * EE + 4 * eq];  // ds_load_b128
            acc[b] *= (C + A * xv);                  // v_pk_fma_f32 + v_pk_mul_f32
        }
    }

    #pragma unroll
    for (int b = 0; b < BT1; ++b) {
        *(v4f*)&hbuf[((size_t)(b0 + b) * HID + h) * EE + 4 * eq] = acc[b];
    }
}

// ---------------- Stage 2 ----------------
// out[b,o,e] = 1 - prod_h ( 1 - dw[h,o,e] * hval[b,h,e] )
// Block: 128 threads. lane -> (o_local in [0,64), e-quad in [0,2)).
// Tile: OT2=64 o, BT2=4 b. h-tile (4*512*8 floats = 64KB) staged async into LDS.
#define OT2 64
#define BT2 4

__global__ __launch_bounds__(128) void dnf_stage2(const float* __restrict__ hbuf,
                                                  const float* __restrict__ dw,
                                                  float* __restrict__ out) {
    __shared__ float hs[BT2 * HID * EE];   // 16384 floats = 64 KB

    const int tid = threadIdx.x;
    const int eq  = tid & 1;
    const int ol  = tid >> 1;         // 0..63
    const int o   = blockIdx.x * OT2 + ol;
    const int b0  = blockIdx.y * BT2;

    // async-stage contiguous 64KB h-tile: hbuf[b0 .. b0+3, :, :]
    stage_64kb_async(hs, hbuf + (size_t)b0 * HID * EE, tid);
    __syncthreads();

    const float* dp = dw + (size_t)o * EE + 4 * eq;

    v4f acc[BT2];
    #pragma unroll
    for (int b = 0; b < BT2; ++b) acc[b] = (v4f){1.0f, 1.0f, 1.0f, 1.0f};

    #pragma unroll 2
    for (int h = 0; h < HID; ++h) {
        v4f d4 = *(const v4f*)(dp + (size_t)h * (OUTF * EE));        // global_load_b128

        #pragma unroll
        for (int b = 0; b < BT2; ++b) {
            v4f hv = *(const v4f*)&hs[(b * HID + h) * EE + 4 * eq];  // ds_load_b128
            acc[b] *= (1.0f - d4 * hv);              // v_pk_fma_f32 + v_pk_mul_f32
        }
    }

    #pragma unroll
    for (int b = 0; b < BT2; ++b) {
        v4f r = 1.0f - acc[b];
        *(v4f*)&out[((size_t)(b0 + b) * OUTF + o) * EE + 4 * eq] = r;  // global_store_b128
    }
}

// ---------------- launcher ----------------
extern "C" void kernel_launch(void* const* d_in, const int* in_sizes, int n_in,
                              void* d_out, int out_size, void* d_ws, size_t ws_size,
                              hipStream_t stream) {
    const float* x  = (const float*)d_in[0];   // (B, IN_F, E)
    const float* cw = (const float*)d_in[1];   // (IN_F, HID_F, E)
    const float* cs = (const float*)d_in[2];   // (IN_F, HID_F, E)
    const float* dw = (const float*)d_in[3];   // (HID_F, OUT_F, E)
    float* out  = (float*)d_out;               // (B, OUT_F, E)
    float* hbuf = (float*)d_ws;                // B*HID_F*E floats = 2 MB scratch

    dim3 g1(HID / HT1, BB / BT1);              // (8, 16) = 128 blocks x 128 thr
    dnf_stage1<<<g1, 128, 0, stream>>>(x, cw, cs, hbuf);

    dim3 g2(OUTF / OT2, BB / BT2);             // (2, 32) = 64 blocks x 128 thr
    dnf_stage2<<<g2, 128, 0, stream>>>(hbuf, dw, out);
}